// BaseModel_28192165331191
// MI455X (gfx1250) — compile-verified
//
#include <hip/hip_runtime.h>
#include <math.h>

// Problem constants (match reference)
#define BINS    100
#define NNODES  10000
#define NCOLS   (NNODES * 2)     // flattened (node, dim) columns = 20000
#define NEVENTS 500000
#define BW      0.01f            // float32(1.0/100)

// Prefix-sum-as-GEMM tiling: Vc padded to 112 rows (rows 100..111 are junk,
// never read) so the WMMA store epilogue is branch-free.
#define MROWS_PAD 112
#define MTILES  7
#define NTILES  (NCOLS / 16)     // 1250
#define TOTAL_TILES (MTILES * NTILES)

typedef float v8f __attribute__((ext_vector_type(8)));
typedef float v2f __attribute__((ext_vector_type(2)));

// ---------------------------------------------------------------------------
// Kernel 1: Vc = L * V via V_WMMA_F32_16X16X4_F32
//   L[m,k] = (k < m) ? BW : 0   (strictly lower triangular, generated in-reg)
//   V[k, col] = v[k, node, dim] with col = node*2 + dim  (row-major [100,20000])
// One wave32 computes one 16x16 output tile; K-loop of 25 steps of K=4.
// ---------------------------------------------------------------------------
__global__ void __launch_bounds__(128)
prefix_cumdisp_wmma(const float* __restrict__ v, float* __restrict__ vc) {
    const int wave = blockIdx.x * (blockDim.x >> 5) + (threadIdx.x >> 5);
    if (wave >= TOTAL_TILES) return;            // uniform per wave -> EXEC all 1s below

    const int lane  = threadIdx.x & 31;
    const int mtile = wave % MTILES;
    const int ntile = wave / MTILES;
    const int colbase = ntile * 16;

    const int n16   = lane & 15;                // col within tile / row-of-A within tile
    const int khalf = (lane >> 4) << 1;         // 0 (lanes 0-15) or 2 (lanes 16-31)
    const int mrow  = mtile * 16 + n16;         // global output row for A operand

    const float* __restrict__ vcol = v + colbase + n16;

    v8f c = {};                                 // f32 accumulator, 8 VGPRs
    for (int kk = 0; kk < BINS; kk += 4) {
        const int kx = kk + khalf;
        v2f a, b;
        // A tile: A[mrow, kx] / A[mrow, kx+1]
        a.x = (kx     < mrow) ? BW : 0.0f;
        a.y = (kx + 1 < mrow) ? BW : 0.0f;
        // B tile: rows kx, kx+1 at columns colbase..colbase+15 (coalesced 64B/row)
        b.x = vcol[(size_t)kx * NCOLS];
        b.y = vcol[(size_t)(kx + 1) * NCOLS];
        // 8 args: (neg_a, A, neg_b, B, c_mod, C, reuse_a, reuse_b)
        c = __builtin_amdgcn_wmma_f32_16x16x4_f32(
                false, a, false, b, (short)0, c, false, false);
    }

    // C/D layout: VGPR r -> row (mbase + r), lane%16 -> col.
    // Vc is padded to 112 rows, so all 8 stores are unconditional.
    const int mbase = mtile * 16 + ((lane >> 4) << 3);  // +0 or +8
    float* __restrict__ outcol = vc + colbase + n16;
#pragma unroll
    for (int r = 0; r < 8; ++r) {
        outcol[(size_t)(mbase + r) * NCOLS] = c[r];
    }
}

// ---------------------------------------------------------------------------
// Kernel 2: per-event intensity.
//   dx = (x0[i]-x0[j]) + (Vc[bin,i]-Vc[bin,j]) + r*(v[bin,i]-v[bin,j])
//   out = exp(-||dx||^2 + beta[i] + beta[j])
// Standardization cancels in the pairwise differences -> use raw x0/v.
// ---------------------------------------------------------------------------
__global__ void __launch_bounds__(256)
event_intensity(const float*  __restrict__ x0,
                const float*  __restrict__ v,
                const float*  __restrict__ beta,
                const float*  __restrict__ times,
                const int*    __restrict__ pairs,
                const float*  __restrict__ vc,
                float*        __restrict__ out) {
    const int e = blockIdx.x * blockDim.x + threadIdx.x;
    if (e >= NEVENTS) return;

    const int   i  = pairs[e];
    const int   j  = pairs[NEVENTS + e];
    const float t  = times[e];

    const float q   = floorf(t / BW);           // jnp.floor_divide
    const float r   = t - q * BW;               // jnp.remainder (positive operands)
    int bin = (int)q;
    bin = bin > (BINS - 1) ? (BINS - 1) : bin;  // clamp bin==100 -> 99

    const float2 xi = *(const float2*)(x0 + 2 * (size_t)i);
    const float2 xj = *(const float2*)(x0 + 2 * (size_t)j);

    const size_t row = (size_t)bin * NCOLS;
    const float2 vi = *(const float2*)(v  + row + 2 * (size_t)i);
    const float2 vj = *(const float2*)(v  + row + 2 * (size_t)j);
    const float2 ci = *(const float2*)(vc + row + 2 * (size_t)i);
    const float2 cj = *(const float2*)(vc + row + 2 * (size_t)j);

    const float dx = (xi.x - xj.x) + (ci.x - cj.x) + r * (vi.x - vj.x);
    const float dy = (xi.y - xj.y) + (ci.y - cj.y) + r * (vi.y - vj.y);

    const float li = -(dx * dx + dy * dy) + beta[i] + beta[j];
    out[e] = __expf(li);
}

// ---------------------------------------------------------------------------
extern "C" void kernel_launch(void* const* d_in, const int* in_sizes, int n_in,
                              void* d_out, int out_size, void* d_ws, size_t ws_size,
                              hipStream_t stream) {
    (void)in_sizes; (void)n_in; (void)out_size; (void)ws_size;

    const float* x0    = (const float*)d_in[0];   // [10000, 2]
    const float* v     = (const float*)d_in[1];   // [100, 10000, 2]
    const float* beta  = (const float*)d_in[2];   // [10000]
    const float* times = (const float*)d_in[3];   // [500000]
    const int*   pairs = (const int*)  d_in[4];   // [2, 500000]
    float*       out   = (float*)d_out;           // [500000]

    float* vc = (float*)d_ws;                     // [112, 20000] = 8.96 MB scratch

    const int wavesPerBlock = 128 / 32;
    const int blocks1 = (TOTAL_TILES + wavesPerBlock - 1) / wavesPerBlock;
    prefix_cumdisp_wmma<<<blocks1, 128, 0, stream>>>(v, vc);

    const int blocks2 = (NEVENTS + 255) / 256;
    event_intensity<<<blocks2, 256, 0, stream>>>(x0, v, beta, times, pairs, vc, out);
}